// PCENetwork_10874857193599
// MI455X (gfx1250) — compile-verified
//
#include <hip/hip_runtime.h>
#include <math.h>

// ---------------- problem constants ----------------
#define BSZ    8
#define HWD    224
#define PS     16
#define NFREQ  4
#define FCH    18            // 2 + 4*NFREQ
#define CH     32
#define NEXP   8
#define NCLS   1000
#define GRPS   8
#define HP     14            // HWD/PS
#define NTOK   (BSZ*HP*HP)   // 1568
#define CAP    245           // int(1.25*NTOK/NEXP)
#define KDIM   (CH*9)        // 288
#define EPSV   1e-5f
#define PIF    3.14159265358979323846f

// padded LDS geometry for the expert kernel
#define XPLANE 336           // padded ci-plane stride (words); rows are 18x18=324 used
#define WROW   34            // padded [co]-row stride (words) for transposed weights

typedef __attribute__((ext_vector_type(2))) float v2f;
typedef __attribute__((ext_vector_type(8))) float v8f;

// ---------------- workspace layout (floats) ----------------
#define O_XBUF   ((size_t)0)                                  // B*CH*H*W
#define O_CHSUM  (O_XBUF + (size_t)BSZ*CH*HWD*HWD)            // 256
#define O_CHSQ   (O_CHSUM + BSZ*CH)                           // 256
#define O_TOP1   (O_CHSQ + BSZ*CH)                            // NTOK ints
#define O_PTOP   (O_TOP1 + NTOK)                              // NTOK
#define O_WN     (O_PTOP + NTOK)                              // NTOK
#define O_STOK   (O_WN + NTOK)                                // NEXP*CAP ints
#define O_SVAL   (O_STOK + NEXP*CAP)                          // NEXP*CAP
#define O_ZSUM   (O_SVAL + NEXP*CAP)                          // 1   (atomic)
#define O_PSUM   (O_ZSUM + 1)                                 // 8   (atomic)
#define O_ECNT   (O_PSUM + NEXP)                              // 8
#define O_FEAT   (O_ECNT + NEXP)                              // 256

__device__ __forceinline__ float silu_f(float v) { return v / (1.f + expf(-v)); }

// ---------------- 1) stem: conv3x3 + folded BN + SiLU ----------------
__global__ void stem_kernel(const float* __restrict__ X, const float* __restrict__ W,
                            const float* __restrict__ bnw, const float* __restrict__ bnb,
                            float* __restrict__ xbuf) {
    size_t idx = (size_t)blockIdx.x * blockDim.x + threadIdx.x;
    if (idx >= (size_t)BSZ*CH*HWD*HWD) return;
    int x = idx % HWD; int y = (idx / HWD) % HWD;
    int co = (idx / (HWD*HWD)) % CH; int b = idx / ((size_t)CH*HWD*HWD);
    float acc = 0.f;
    #pragma unroll
    for (int ci = 0; ci < 3; ++ci)
        #pragma unroll
        for (int kh = 0; kh < 3; ++kh) {
            int sy = y + kh - 1;
            if ((unsigned)sy >= (unsigned)HWD) continue;
            #pragma unroll
            for (int kw = 0; kw < 3; ++kw) {
                int sx = x + kw - 1;
                if ((unsigned)sx >= (unsigned)HWD) continue;
                acc += X[(((size_t)b*3 + ci)*HWD + sy)*HWD + sx] *
                       W[((co*3 + ci)*3 + kh)*3 + kw];
            }
        }
    float v = acc * bnw[co] + bnb[co];
    xbuf[idx] = silu_f(v);
}

// ---------------- 2) router gate (one block per token) ----------------
__global__ void gate_kernel(const float* __restrict__ xbuf,
                            const float* __restrict__ gw, const float* __restrict__ gb,
                            float* __restrict__ ws) {
    __shared__ float means[CH + FCH];
    __shared__ float logits[NEXP];
    int n = blockIdx.x;
    int b = n / (HP*HP); int rem = n % (HP*HP);
    int hi = rem / HP, wi = rem % HP;
    int t = threadIdx.x;

    if (t < CH) {                 // channel means over the 16x16 patch
        const float* base = xbuf + (((size_t)b*CH + t)*HWD + hi*PS)*HWD + wi*PS;
        float s = 0.f;
        for (int py = 0; py < PS; ++py)
            for (int px = 0; px < PS; ++px) s += base[(size_t)py*HWD + px];
        means[t] = s * (1.f/(PS*PS));
    } else if (t < CH + FCH) {    // analytic Fourier-feature patch means
        int j = t - CH;
        float s = 0.f;
        if (j == 0) { for (int p = 0; p < PS; ++p) s += (hi*PS + p + 0.5f)/HWD; }
        else if (j == 1) { for (int p = 0; p < PS; ++p) s += (wi*PS + p + 0.5f)/HWD; }
        else {
            int k = (j - 2) / 4, r = (j - 2) % 4;
            float f = (float)(1 << k) * PIF;
            int base = (r < 2) ? hi : wi;
            for (int p = 0; p < PS; ++p) {
                float c = (base*PS + p + 0.5f)/HWD;
                s += (r == 0 || r == 2) ? sinf(f*c) : cosf(f*c);
            }
        }
        means[t] = s * (1.f/PS);
    }
    __syncthreads();

    if (t < NEXP) {
        float acc = gb[t];
        for (int j = 0; j < CH + FCH; ++j) acc += means[j] * gw[j*NEXP + t];
        logits[t] = acc;          // TEMP == 1.0
    }
    __syncthreads();

    if (t == 0) {
        float mx = logits[0]; int am = 0;
        for (int e = 1; e < NEXP; ++e) if (logits[e] > mx) { mx = logits[e]; am = e; }
        float se = 0.f, pe[NEXP];
        for (int e = 0; e < NEXP; ++e) { pe[e] = expf(logits[e] - mx); se += pe[e]; }
        float lse = mx + logf(se);
        ((int*)(ws + O_TOP1))[n] = am;
        (ws + O_PTOP)[n] = pe[am] / se;
        atomicAdd(ws + O_ZSUM, lse*lse);
        for (int e = 0; e < NEXP; ++e) atomicAdd(ws + O_PSUM + e, pe[e]/se);
    }
}

// ---------------- 3) capacity-limited top-1 routing scan ----------------
__global__ void route_kernel(float* __restrict__ ws) {
    int e = threadIdx.x;
    if (e >= NEXP) return;
    const int*  top1 = (const int*)(ws + O_TOP1);
    int*   stok = (int*)(ws + O_STOK);
    float* sval = ws + O_SVAL;
    float* wn   = ws + O_WN;
    const float* pt = ws + O_PTOP;
    int cnt = 0;
    for (int n = 0; n < NTOK; ++n) {
        if (top1[n] == e) {
            if (cnt < CAP) { stok[e*CAP + cnt] = n; sval[e*CAP + cnt] = 1.f; wn[n] = pt[n]; }
            else           { wn[n] = 0.f; }
            ++cnt;
        }
    }
    for (int s = (cnt < CAP ? cnt : CAP); s < CAP; ++s) { stok[e*CAP + s] = 0; sval[e*CAP + s] = 0.f; }
    (ws + O_ECNT)[e] = (float)cnt;
}

// ---------------- 4) expert conv3x3 via V_WMMA_F32_16X16X4_F32 ----------------
// One block per (expert, slot); 8 waves. Tap-major implicit GEMM:
//   y[co, py*16+px] = sum_{kh,kw} sum_ci W[r][co][ci] * Xpad[ci][py+kh][px+kw]
// Xpad zero-padded to 18x18 in LDS -> no boundary checks, no EXEC divergence.
// Per (tile, tap) the B address is base + ci*XPLANE: straight-line 72 WMMAs/tile.
__global__ void expert_kernel(float* __restrict__ xbuf,
                              const float* __restrict__ ew,   // (E,CH,CH,3,3)
                              const float* __restrict__ eb,   // (E,CH)
                              const float* __restrict__ gamma,
                              const float* __restrict__ ws) {
    __shared__ float x_sp[CH * XPLANE];        // padded token: [ci][18][18] (+pad)
    __shared__ float w_t[9 * CH * WROW];       // transposed weights: [r][co][ci(+pad)]
    int e = blockIdx.x / CAP, s = blockIdx.x % CAP;
    float val = (ws + O_SVAL)[e*CAP + s];
    if (val == 0.f) return;                    // block-uniform
    int tok = ((const int*)(ws + O_STOK))[e*CAP + s];
    float wn = (ws + O_WN)[tok];
    int b = tok / (HP*HP); int rem = tok % (HP*HP);
    int hi = rem / HP, wi = rem % HP;
    int tid = threadIdx.x;

    // padded token load: 32*18 = 576 rows of 18 (rows 0/17 and cols 0/17 zero)
    for (int rr = tid; rr < CH*18; rr += 256) {
        int ci = rr / 18, syp = rr % 18;
        float* dst = x_sp + ci*XPLANE + syp*18;
        if (syp == 0 || syp == 17) {
            for (int j = 0; j < 18; ++j) dst[j] = 0.f;
        } else {
            const float* src = xbuf + (((size_t)b*CH + ci)*HWD + hi*PS + (syp-1))*HWD + wi*PS;
            dst[0] = 0.f; dst[17] = 0.f;
            for (int px = 0; px < PS; ++px) dst[px+1] = src[px];
        }
    }
    // weight transpose: src i = co*288 + ci*9 + r  ->  w_t[(r*CH + co)*WROW + ci]
    for (int i = tid; i < CH*KDIM; i += 256) {
        int co = i / KDIM, remi = i % KDIM, ci = remi / 9, r = remi % 9;
        w_t[(r*CH + co)*WROW + ci] = ew[(size_t)e*CH*KDIM + i];
    }
    __syncthreads();

    int wave = tid >> 5, lane = tid & 31;
    int mA = lane & 15;                        // A row (M) / B col (N) for this lane
    int khalf = (lane >> 4) << 1;              // lanes 16-31 hold K+2,K+3

    for (int t = wave; t < 32; t += 8) {       // 2 co_tiles x 16 n_tiles
        int co_tile = t >> 4, n_tile = t & 15; // py == n_tile (wave-uniform), px = mA
        v8f acc = {};
        #pragma unroll
        for (int kh = 0; kh < 3; ++kh) {
            #pragma unroll
            for (int kw = 0; kw < 3; ++kw) {
                int r = kh*3 + kw;
                const float* wrow = w_t + ((r*CH + co_tile*16 + mA)*WROW + khalf);
                const float* bcol = x_sp + (khalf*XPLANE + (n_tile + kh)*18 + mA + kw);
                #pragma unroll
                for (int c0 = 0; c0 < CH; c0 += 4) {
                    v2f a, bv;
                    a.x  = wrow[c0];
                    a.y  = wrow[c0 + 1];
                    bv.x = bcol[c0 * XPLANE];
                    bv.y = bcol[(c0 + 1) * XPLANE];
                    acc = __builtin_amdgcn_wmma_f32_16x16x4_f32(
                              false, a, false, bv, (short)0, acc, false, false);
                }
            }
        }
        #pragma unroll
        for (int i = 0; i < 8; ++i) {          // C/D layout: vgpr i -> rows i / i+8
            int co = co_tile*16 + ((lane >> 4) ? (8 + i) : i);
            float v = acc[i] + eb[e*CH + co];
            float y = silu_f(v);
            size_t oi = (((size_t)b*CH + co)*HWD + hi*PS + n_tile)*HWD + wi*PS + mA;
            xbuf[oi] += gamma[co] * wn * y;    // residual add in place (token-exclusive)
        }
    }
}

// ---------------- 5) per-(b,c) sum / sumsq reduction ----------------
__global__ void reduce_kernel(const float* __restrict__ xbuf, float* __restrict__ ws) {
    __shared__ float ss[256], sq[256];
    int bc = blockIdx.x;                // 0..255
    const float* base = xbuf + (size_t)bc*HWD*HWD;
    float a = 0.f, a2 = 0.f;
    for (int i = threadIdx.x; i < HWD*HWD; i += 256) {
        float v = base[i]; a += v; a2 += v*v;
    }
    ss[threadIdx.x] = a; sq[threadIdx.x] = a2;
    __syncthreads();
    for (int o = 128; o > 0; o >>= 1) {
        if (threadIdx.x < o) { ss[threadIdx.x] += ss[threadIdx.x+o]; sq[threadIdx.x] += sq[threadIdx.x+o]; }
        __syncthreads();
    }
    if (threadIdx.x == 0) { (ws + O_CHSUM)[bc] = ss[0]; (ws + O_CHSQ)[bc] = sq[0]; }
}

// ---------------- 6) GN(algebraic) -> pool -> LN -> losses ----------------
__global__ void finalize_kernel(const float* __restrict__ gnw, const float* __restrict__ gnb,
                                const float* __restrict__ lnw, const float* __restrict__ lnb,
                                float* __restrict__ ws, float* __restrict__ out) {
    __shared__ float gm[BSZ*GRPS], gv[BSZ*GRPS], ft[BSZ*CH], mu[BSZ], vr[BSZ];
    int t = threadIdx.x;
    const float hw = (float)(HWD*HWD);
    if (t < BSZ*GRPS) {
        int b = t / GRPS, g = t % GRPS;
        float s = 0.f, s2 = 0.f;
        for (int c = g*(CH/GRPS); c < (g+1)*(CH/GRPS); ++c) {
            s += (ws + O_CHSUM)[b*CH + c]; s2 += (ws + O_CHSQ)[b*CH + c];
        }
        float cnt = (CH/GRPS) * hw;
        float m = s / cnt;
        gm[t] = m; gv[t] = s2/cnt - m*m;
    }
    __syncthreads();
    {
        int b = t / CH, c = t % CH, g = c / (CH/GRPS);
        float pooled = (ws + O_CHSUM)[t] / hw;
        ft[t] = gnw[c] * (pooled - gm[b*GRPS + g]) * rsqrtf(gv[b*GRPS + g] + EPSV) + gnb[c];
    }
    __syncthreads();
    if (t < BSZ) {
        float m = 0.f;
        for (int c = 0; c < CH; ++c) m += ft[t*CH + c];
        m /= CH;
        float v = 0.f;
        for (int c = 0; c < CH; ++c) { float d = ft[t*CH + c] - m; v += d*d; }
        mu[t] = m; vr[t] = v / CH;
    }
    __syncthreads();
    {
        int b = t / CH, c = t % CH;
        (ws + O_FEAT)[t] = (ft[t] - mu[b]) * rsqrtf(vr[b] + EPSV) * lnw[c] + lnb[c];
    }
    if (t == 0) {
        out[BSZ*NCLS] = (ws + O_ZSUM)[0] / (float)NTOK;           // z_loss
        float imb = 0.f;
        for (int e = 0; e < NEXP; ++e)
            imb += ((ws + O_PSUM)[e] / NTOK) * ((ws + O_ECNT)[e] / NTOK);
        out[BSZ*NCLS + 1] = (float)NEXP * imb;                    // imb_loss
    }
}

// ---------------- 7) classifier head ----------------
__global__ void head_kernel(const float* __restrict__ ws, const float* __restrict__ hw,
                            const float* __restrict__ hb, float* __restrict__ out) {
    int idx = blockIdx.x * blockDim.x + threadIdx.x;
    if (idx >= BSZ*NCLS) return;
    int b = idx / NCLS, k = idx % NCLS;
    float acc = hb[k];
    #pragma unroll
    for (int c = 0; c < CH; ++c) acc += (ws + O_FEAT)[b*CH + c] * hw[c*NCLS + k];
    out[idx] = acc;
}

// ---------------- launcher ----------------
extern "C" void kernel_launch(void* const* d_in, const int* in_sizes, int n_in,
                              void* d_out, int out_size, void* d_ws, size_t ws_size,
                              hipStream_t stream) {
    const float* X      = (const float*)d_in[0];
    const float* stem_w = (const float*)d_in[1];
    const float* bn_w   = (const float*)d_in[2];
    const float* bn_b   = (const float*)d_in[3];
    const float* gate_w = (const float*)d_in[4];
    const float* gate_b = (const float*)d_in[5];
    const float* exp_w  = (const float*)d_in[6];
    const float* exp_b  = (const float*)d_in[7];
    const float* gamma  = (const float*)d_in[8];
    const float* gn_w   = (const float*)d_in[9];
    const float* gn_b   = (const float*)d_in[10];
    const float* ln_w   = (const float*)d_in[11];
    const float* ln_b   = (const float*)d_in[12];
    const float* head_w = (const float*)d_in[13];
    const float* head_b = (const float*)d_in[14];
    float* ws  = (float*)d_ws;
    float* out = (float*)d_out;

    hipMemsetAsync(ws + O_ZSUM, 0, (1 + NEXP) * sizeof(float), stream);

    {   size_t tot = (size_t)BSZ*CH*HWD*HWD;
        stem_kernel<<<(unsigned)((tot + 255)/256), 256, 0, stream>>>(X, stem_w, bn_w, bn_b, ws + O_XBUF); }
    gate_kernel<<<NTOK, 64, 0, stream>>>(ws + O_XBUF, gate_w, gate_b, ws);
    route_kernel<<<1, 32, 0, stream>>>(ws);
    expert_kernel<<<NEXP*CAP, 256, 0, stream>>>(ws + O_XBUF, exp_w, exp_b, gamma, ws);
    reduce_kernel<<<BSZ*CH, 256, 0, stream>>>(ws + O_XBUF, ws);
    finalize_kernel<<<1, BSZ*CH, 0, stream>>>(gn_w, gn_b, ln_w, ln_b, ws, out);
    head_kernel<<<(BSZ*NCLS + 255)/256, 256, 0, stream>>>(ws, head_w, head_b, out);
}